// MAGNN_layer_42820823941456
// MI455X (gfx1250) — compile-verified
//
#include <hip/hip_runtime.h>
#include <math.h>

#define Hh 8      // heads
#define Ff 32     // feats per head
#define Dd 256    // node dim

typedef float v2f __attribute__((ext_vector_type(2)));
typedef float v4f __attribute__((ext_vector_type(4)));
typedef float v8f __attribute__((ext_vector_type(8)));

// ---------- CDNA5 async global->LDS copy (ASYNCcnt) ----------
__device__ __forceinline__ void async_copy_b128(void* lds, const float* g) {
    unsigned l = (unsigned)(uintptr_t)lds;   // low 32 bits of generic LDS ptr = LDS offset
    asm volatile("global_load_async_to_lds_b128 %0, %1, off"
                 :: "v"(l), "v"(g)
                 : "memory");
}
__device__ __forceinline__ void wait_async0() {
    asm volatile("s_wait_asynccnt 0x0" ::: "memory");
}

// ---------- order-preserving float<->uint for atomicMax over possibly-negative values
__device__ __forceinline__ unsigned ordEnc(float f) {
    unsigned u = __float_as_uint(f);
    return (u & 0x80000000u) ? ~u : (u | 0x80000000u);
}
__device__ __forceinline__ float ordDec(unsigned u) {
    return (u & 0x80000000u) ? __uint_as_float(u & 0x7FFFFFFFu) : __uint_as_float(~u);
}

// ---------- fills
__global__ void fill_f32(float* p, float v, long long n) {
    long long i = (long long)blockIdx.x * blockDim.x + threadIdx.x;
    if (i < n) p[i] = v;
}
__global__ void fill_u32(unsigned* p, unsigned v, long long n) {
    long long i = (long long)blockIdx.x * blockDim.x + threadIdx.x;
    if (i < n) p[i] = v;
}

// ---------- edge pass 1: metapath feature -> attention logit + segment max
// one thread per (edge, head); all row reads vectorized to b128
template <int NP>
__global__ void edge_logits(const float* __restrict__ fA, const float* __restrict__ fB,
                            const int* __restrict__ idx, const float* __restrict__ attn_r,
                            const float* __restrict__ r_ab, const float* __restrict__ r_ba,
                            float* __restrict__ e_buf, unsigned* __restrict__ m_buf, int E) {
    int t = blockIdx.x * blockDim.x + threadIdx.x;
    if (t >= E * Hh) return;
    int e = t >> 3, head = t & 7;
    const int* ip = idx + e * NP;
    int dst = ip[0];
    int base = head * Ff;
    const v4f* f0v = (const v4f*)(fA + (size_t)dst * Dd + base);
    const v4f* f1v = (const v4f*)(fB + (size_t)ip[1] * Dd + base);
    const v4f* f2v = (NP == 3) ? (const v4f*)(fA + (size_t)ip[2] * Dd + base) : f0v;
    const v4f* rv4 = (const v4f*)(r_ab + base);
    const v4f* sv4 = (const v4f*)(r_ba + base);
    const v4f* av4 = (const v4f*)(attn_r + base);
    const float inv = (NP == 3) ? (1.0f / 3.0f) : 0.5f;
    float er = 0.0f;
#pragma unroll
    for (int q = 0; q < Ff / 4; ++q) {           // 2 complex pairs per iteration
        v4f x0 = f0v[q], x1 = f1v[q], rv = rv4[q];
        float h_re  = x0[0] + x1[0] * rv[0] - x1[1] * rv[1];
        float h_im  = x0[1] + x1[0] * rv[1] + x1[1] * rv[0];
        float h_re2 = x0[2] + x1[2] * rv[2] - x1[3] * rv[3];
        float h_im2 = x0[3] + x1[2] * rv[3] + x1[3] * rv[2];
        if (NP == 3) {
            v4f sv = sv4[q], x2 = f2v[q];
            float t_re  = rv[0] * sv[0] - rv[1] * sv[1];
            float t_im  = rv[0] * sv[1] + rv[1] * sv[0];
            float t_re2 = rv[2] * sv[2] - rv[3] * sv[3];
            float t_im2 = rv[2] * sv[3] + rv[3] * sv[2];
            h_re  += x2[0] * t_re  - x2[1] * t_im;
            h_im  += x2[0] * t_im  + x2[1] * t_re;
            h_re2 += x2[2] * t_re2 - x2[3] * t_im2;
            h_im2 += x2[2] * t_im2 + x2[3] * t_re2;
        }
        v4f av = av4[q];
        er += h_re * av[0] + h_im * av[1] + h_re2 * av[2] + h_im2 * av[3];
    }
    er *= inv;
    float lr = (er > 0.0f) ? er : 0.01f * er;    // LeakyReLU(0.01)
    e_buf[t] = lr;
    atomicMax(m_buf + (size_t)dst * Hh + head, ordEnc(lr));
}

// ---------- edge pass 2: exp(e - max) + segment sum
template <int NP>
__global__ void edge_exp(const int* __restrict__ idx, const unsigned* __restrict__ m_buf,
                         float* __restrict__ e_buf, float* __restrict__ s_buf, int E) {
    int t = blockIdx.x * blockDim.x + threadIdx.x;
    if (t >= E * Hh) return;
    int e = t >> 3, head = t & 7;
    int dst = idx[e * NP];
    float m = ordDec(m_buf[(size_t)dst * Hh + head]);
    float ex = expf(e_buf[t] - m);
    e_buf[t] = ex;
    atomicAdd(s_buf + (size_t)dst * Hh + head, ex);
}

// ---------- edge pass 3: recompute metapath feature, scatter h * (ex/sum)
template <int NP>
__global__ void edge_scatter(const float* __restrict__ fA, const float* __restrict__ fB,
                             const int* __restrict__ idx, const float* __restrict__ r_ab,
                             const float* __restrict__ r_ba, const float* __restrict__ e_buf,
                             const float* __restrict__ s_buf, float* __restrict__ out, int E) {
    int t = blockIdx.x * blockDim.x + threadIdx.x;
    if (t >= E * Hh) return;
    int e = t >> 3, head = t & 7;
    const int* ip = idx + e * NP;
    int dst = ip[0];
    float a = e_buf[t] / s_buf[(size_t)dst * Hh + head];
    int base = head * Ff;
    const v4f* f0v = (const v4f*)(fA + (size_t)dst * Dd + base);
    const v4f* f1v = (const v4f*)(fB + (size_t)ip[1] * Dd + base);
    const v4f* f2v = (NP == 3) ? (const v4f*)(fA + (size_t)ip[2] * Dd + base) : f0v;
    const v4f* rv4 = (const v4f*)(r_ab + base);
    const v4f* sv4 = (const v4f*)(r_ba + base);
    const float s = ((NP == 3) ? (1.0f / 3.0f) : 0.5f) * a;
    float* od = out + (size_t)dst * Dd + base;
#pragma unroll
    for (int q = 0; q < Ff / 4; ++q) {
        v4f x0 = f0v[q], x1 = f1v[q], rv = rv4[q];
        float h_re  = x0[0] + x1[0] * rv[0] - x1[1] * rv[1];
        float h_im  = x0[1] + x1[0] * rv[1] + x1[1] * rv[0];
        float h_re2 = x0[2] + x1[2] * rv[2] - x1[3] * rv[3];
        float h_im2 = x0[3] + x1[2] * rv[3] + x1[3] * rv[2];
        if (NP == 3) {
            v4f sv = sv4[q], x2 = f2v[q];
            float t_re  = rv[0] * sv[0] - rv[1] * sv[1];
            float t_im  = rv[0] * sv[1] + rv[1] * sv[0];
            float t_re2 = rv[2] * sv[2] - rv[3] * sv[3];
            float t_im2 = rv[2] * sv[3] + rv[3] * sv[2];
            h_re  += x2[0] * t_re  - x2[1] * t_im;
            h_im  += x2[0] * t_im  + x2[1] * t_re;
            h_re2 += x2[2] * t_re2 - x2[3] * t_im2;
            h_im2 += x2[2] * t_im2 + x2[3] * t_re2;
        }
        int d = q * 4;
        atomicAdd(od + d,     h_re  * s);
        atomicAdd(od + d + 1, h_im  * s);
        atomicAdd(od + d + 2, h_re2 * s);
        atomicAdd(od + d + 3, h_im2 * s);
    }
}

// ---------- ELU in place
__global__ void elu_inplace(float* __restrict__ x, long long n) {
    long long i = (long long)blockIdx.x * blockDim.x + threadIdx.x;
    if (i < n) {
        float v = x[i];
        x[i] = (v > 0.0f) ? v : expm1f(v);
    }
}

// =======================================================================
// WMMA fp32 GEMM, LDS-staged via async global->LDS copies.
// Block = 256 threads (8 waves). Block tile: 16 rows x 128 cols.
// K-chunks of 64 staged in LDS: A 16x64 (1 b128/thread), B 64x128 (8 b128/thread).
// Each wave computes one 16x16 tile (wave id selects column tile).
// grid = (rows/16, Ncols/128).
// =======================================================================
#define SA_STRIDE 68    // 272B rows: 16B aligned, bank-spread
#define SB_STRIDE 132   // 528B rows: 16B aligned, bank-spread

template <bool SCORE>
__global__ void wmma_gemm_lds(const float* __restrict__ X, const float* __restrict__ W,
                              const float* __restrict__ bias, const float* __restrict__ vvec,
                              float* __restrict__ OutOrScore, int K, int Ncols) {
    __shared__ __align__(16) float sA[16 * SA_STRIDE];
    __shared__ __align__(16) float sB[64 * SB_STRIDE];

    const int row0  = blockIdx.x * 16;
    const int colB0 = blockIdx.y * 128;
    const int lane  = threadIdx.x & 31;
    const int wid   = threadIdx.x >> 5;      // 0..7 -> column tile
    const int mlo   = lane & 15;
    const int khalf = lane >> 4;             // K pair select for fp32 fragments
    const int colw  = wid * 16;

    v8f acc = {};
    for (int kt = 0; kt < K; kt += 64) {
        __syncthreads();                     // previous chunk fully consumed
        {   // A chunk: 16 rows x 64 floats = 256 b128 segments
            int rowa = threadIdx.x >> 4;                 // 0..15
            int sega = (threadIdx.x & 15) << 2;          // 0,4,...,60
            async_copy_b128(&sA[rowa * SA_STRIDE + sega],
                            X + (size_t)(row0 + rowa) * K + kt + sega);
        }
#pragma unroll
        for (int i = 0; i < 8; ++i) {        // B chunk: 64 rows x 128 floats = 2048 segs
            int idxb = threadIdx.x + i * 256;
            int rowb = idxb >> 5;                        // 0..63
            int segb = (idxb & 31) << 2;                 // 0..124
            async_copy_b128(&sB[rowb * SB_STRIDE + segb],
                            W + (size_t)(kt + rowb) * Ncols + colB0 + segb);
        }
        wait_async0();
        __syncthreads();                     // all async writes visible block-wide
#pragma unroll
        for (int kk = 0; kk < 64; kk += 4) {
            int ka = kk + 2 * khalf;
            v2f a = *(const v2f*)&sA[mlo * SA_STRIDE + ka];
            v2f b;
            b[0] = sB[ka * SB_STRIDE + colw + mlo];
            b[1] = sB[(ka + 1) * SB_STRIDE + colw + mlo];
            acc = __builtin_amdgcn_wmma_f32_16x16x4_f32(false, a, false, b, (short)0, acc,
                                                        false, false);
        }
    }

    const int col = colB0 + colw + mlo;
    const float bv = bias[col];
    if (SCORE) {
        const float vv = vvec[col];
        float partial = 0.0f;
#pragma unroll
        for (int i = 0; i < 8; ++i) partial += tanhf(acc[i] + bv) * vv;
        for (int off = 16; off > 0; off >>= 1) partial += __shfl_down(partial, off, 32);
        if (lane == 0) atomicAdd(OutOrScore, partial);
    } else {
        const int mhi = khalf * 8;           // C/D layout: vgpr i holds M = i + 8*(lane/16)
#pragma unroll
        for (int i = 0; i < 8; ++i)
            OutOrScore[(size_t)(row0 + i + mhi) * Ncols + col] = acc[i] + bv;
    }
}

// ---------- beta = softmax([s_aba, s_ab] / N)
__global__ void beta_kernel(const float* __restrict__ score, float* __restrict__ beta,
                            float invN) {
    float s0 = score[0] * invN, s1 = score[1] * invN;
    float m = fmaxf(s0, s1);
    float e0 = expf(s0 - m), e1 = expf(s1 - m);
    float d = e0 + e1;
    beta[0] = e0 / d;
    beta[1] = e1 / d;
}

// ---------- inter_A = b0*x_aba + b1*x_ab (in place); inter_B = feat_B
__global__ void combine(float* __restrict__ interA, const float* __restrict__ x_ab,
                        const float* __restrict__ fB, float* __restrict__ interB,
                        const float* __restrict__ beta, long long n) {
    long long i = (long long)blockIdx.x * blockDim.x + threadIdx.x;
    if (i < n) {
        interA[i] = beta[0] * interA[i] + beta[1] * x_ab[i];
        interB[i] = fB[i];
    }
}

extern "C" void kernel_launch(void* const* d_in, const int* in_sizes, int n_in,
                              void* d_out, int out_size, void* d_ws, size_t ws_size,
                              hipStream_t stream) {
    const float* feat_A   = (const float*)d_in[0];
    const float* feat_B   = (const float*)d_in[1];
    const float* attn_aba = (const float*)d_in[2];
    const float* attn_ab  = (const float*)d_in[3];
    const float* r_ab     = (const float*)d_in[4];
    const float* r_ba     = (const float*)d_in[5];
    const float* W_inter  = (const float*)d_in[6];
    const float* b_inter  = (const float*)d_in[7];
    const float* attn_vec = (const float*)d_in[8];
    const float* W_out    = (const float*)d_in[9];
    const float* b_out    = (const float*)d_in[10];
    const int*   idx_aba  = (const int*)d_in[11];
    const int*   idx_ab   = (const int*)d_in[12];

    const int N     = in_sizes[0] / Dd;      // 50000 (multiple of 16)
    const int E_aba = in_sizes[11] / 3;      // 200000
    const int E_ab  = in_sizes[12] / 2;      // 200000
    const long long ND = (long long)N * Dd;

    // output slices: [final_A, final_B, inter_A, inter_B]
    float* finalA = (float*)d_out;
    float* finalB = finalA + ND;
    float* interA = finalA + 2 * ND;
    float* interB = finalA + 3 * ND;

    // workspace layout (floats): x_ab | e_buf | m_buf(u32) | s_buf | score[2] | beta[2]
    float*    x_ab  = (float*)d_ws;
    float*    e_buf = x_ab + ND;                                // E*H
    unsigned* m_buf = (unsigned*)(e_buf + (long long)E_aba * Hh);
    float*    s_buf = (float*)(m_buf + (long long)N * Hh);
    float*    score = s_buf + (long long)N * Hh;
    float*    beta  = score + 2;

    const int TB = 256;
    const long long NH = (long long)N * Hh;
    dim3 bGemm(256, 1, 1);

    auto blocks = [](long long n, int tb) { return (unsigned)((n + tb - 1) / tb); };

    // ===== metapath A-B-A (accumulate into interA slice) =====
    fill_u32<<<blocks(NH, TB), TB, 0, stream>>>(m_buf, 0u, NH);
    fill_f32<<<blocks(NH, TB), TB, 0, stream>>>(s_buf, 0.0f, NH);
    fill_f32<<<blocks(ND, TB), TB, 0, stream>>>(interA, 0.0f, ND);
    fill_f32<<<1, 32, 0, stream>>>(score, 0.0f, 2);

    long long EH3 = (long long)E_aba * Hh;
    edge_logits<3><<<blocks(EH3, TB), TB, 0, stream>>>(feat_A, feat_B, idx_aba, attn_aba,
                                                       r_ab, r_ba, e_buf, m_buf, E_aba);
    edge_exp<3><<<blocks(EH3, TB), TB, 0, stream>>>(idx_aba, m_buf, e_buf, s_buf, E_aba);
    edge_scatter<3><<<blocks(EH3, TB), TB, 0, stream>>>(feat_A, feat_B, idx_aba, r_ab, r_ba,
                                                        e_buf, s_buf, interA, E_aba);
    elu_inplace<<<blocks(ND, TB), TB, 0, stream>>>(interA, ND);

    // ===== metapath A-B (accumulate into workspace x_ab) =====
    fill_u32<<<blocks(NH, TB), TB, 0, stream>>>(m_buf, 0u, NH);
    fill_f32<<<blocks(NH, TB), TB, 0, stream>>>(s_buf, 0.0f, NH);
    fill_f32<<<blocks(ND, TB), TB, 0, stream>>>(x_ab, 0.0f, ND);

    long long EH2 = (long long)E_ab * Hh;
    edge_logits<2><<<blocks(EH2, TB), TB, 0, stream>>>(feat_A, feat_B, idx_ab, attn_ab,
                                                       r_ab, r_ba, e_buf, m_buf, E_ab);
    edge_exp<2><<<blocks(EH2, TB), TB, 0, stream>>>(idx_ab, m_buf, e_buf, s_buf, E_ab);
    edge_scatter<2><<<blocks(EH2, TB), TB, 0, stream>>>(feat_A, feat_B, idx_ab, r_ab, r_ba,
                                                        e_buf, s_buf, x_ab, E_ab);
    elu_inplace<<<blocks(ND, TB), TB, 0, stream>>>(x_ab, ND);

    // ===== inter-metapath attention scores (WMMA f32 + async-LDS staging) =====
    dim3 gScore(N / 16, 128 / 128, 1);
    wmma_gemm_lds<true><<<gScore, bGemm, 0, stream>>>(interA, W_inter, b_inter, attn_vec,
                                                      score + 0, Dd, 128);
    wmma_gemm_lds<true><<<gScore, bGemm, 0, stream>>>(x_ab, W_inter, b_inter, attn_vec,
                                                      score + 1, Dd, 128);
    beta_kernel<<<1, 1, 0, stream>>>(score, beta, 1.0f / (float)N);

    // ===== combine: inter_A (in place), inter_B = feat_B =====
    combine<<<blocks(ND, TB), TB, 0, stream>>>(interA, x_ab, feat_B, interB, beta, ND);

    // ===== output projection (WMMA f32): final_A, final_B =====
    dim3 gOut(N / 16, Dd / 128, 1);
    wmma_gemm_lds<false><<<gOut, bGemm, 0, stream>>>(interA, W_out, b_out, nullptr,
                                                     finalA, Dd, Dd);
    wmma_gemm_lds<false><<<gOut, bGemm, 0, stream>>>(feat_B, W_out, b_out, nullptr,
                                                     finalB, Dd, Dd);
}